// GenerativeUpsample_82944408420602
// MI455X (gfx1250) — compile-verified
//
#include <hip/hip_runtime.h>
#include <cstdint>

#define BATCHES 8
#define FEAC 64

// ---- workspace layout (uint32 indices) ----
#define WS_START   0            // B+1 segment boundaries
#define WS_COUNT   16           // per-batch counts
#define WS_KREM    24           // remaining rank within candidate set
#define WS_PREFIX  32           // resolved high bits of ordered-uint kth value
#define WS_ACTIVE  40           // counts > target ?
#define WS_THRESH  48           // final per-batch float threshold
#define WS_HIST    64           // 8 batches x 256 bins
#define WS_TOTAL   (64 + BATCHES * 256)

__device__ __forceinline__ unsigned f2ord(float x) {
  unsigned u = __float_as_uint(x);
  return (u & 0x80000000u) ? ~u : (u | 0x80000000u);
}
__device__ __forceinline__ float ord2f(unsigned u) {
  unsigned b = (u & 0x80000000u) ? (u & 0x7FFFFFFFu) : ~u;
  return __uint_as_float(b);
}

// ---- zero the control workspace ----
__global__ void k_init(unsigned* __restrict__ ws) {
  for (int i = threadIdx.x; i < WS_TOTAL; i += blockDim.x) ws[i] = 0u;
}

// ---- batch_ids is sorted: find segment boundaries, no atomics ----
__global__ void k_boundary(const int* __restrict__ bids, int n, int* __restrict__ ws) {
  int i = blockIdx.x * blockDim.x + threadIdx.x;
  if (i >= n) return;
  int a = bids[i];
  int pa = (i == 0) ? -1 : bids[i - 1];
  for (int b = pa + 1; b <= a; ++b) ws[WS_START + b] = i;
  if (i == n - 1)
    for (int b = a + 1; b <= BATCHES; ++b) ws[WS_START + b] = n;
}

// ---- counts, activity, initial rank ----
__global__ void k_setup(const int* __restrict__ target, unsigned* __restrict__ ws) {
  int t = threadIdx.x;
  if (t < BATCHES) {
    int* iws = (int*)ws;
    int cnt = iws[WS_START + t + 1] - iws[WS_START + t];
    iws[WS_COUNT + t] = cnt;
    int tg = target[t];
    int act = cnt > tg;
    iws[WS_ACTIVE + t] = act;
    iws[WS_KREM + t] = act ? (cnt - tg - 1) : 0;
    ws[WS_PREFIX + t] = 0u;
    if (!act) ((float*)ws)[WS_THRESH + t] = -3.4028234663852886e38f; // finfo(f32).min
  }
}

// ---- one radix pass: histogram 8-bit digit of candidates per batch ----
__global__ void k_hist(const float* __restrict__ pred, const int* __restrict__ bids,
                       int n, unsigned* __restrict__ ws, int shift) {
  __shared__ unsigned lh[BATCHES * 256];
  __shared__ unsigned pref[BATCHES];
  __shared__ int act[BATCHES];
  for (int i = threadIdx.x; i < BATCHES * 256; i += blockDim.x) lh[i] = 0u;
  if (threadIdx.x < BATCHES) {
    pref[threadIdx.x] = ws[WS_PREFIX + threadIdx.x];
    act[threadIdx.x]  = (int)ws[WS_ACTIVE + threadIdx.x];
  }
  __syncthreads();
  int stride = gridDim.x * blockDim.x;
  for (int i = blockIdx.x * blockDim.x + threadIdx.x; i < n; i += stride) {
    int b = bids[i];
    if (!act[b]) continue;
    unsigned u = f2ord(pred[i]);
    if (shift != 24 && (((u ^ pref[b]) >> (shift + 8)) != 0u)) continue;
    atomicAdd(&lh[(b << 8) + ((u >> shift) & 255u)], 1u);
  }
  __syncthreads();
  for (int i = threadIdx.x; i < BATCHES * 256; i += blockDim.x)
    if (lh[i]) atomicAdd(&ws[WS_HIST + i], lh[i]);
}

// ---- pick digit per batch via 256-wide LDS scan; zero hist for next pass ----
__global__ void k_select(unsigned* __restrict__ ws, int shift) {
  __shared__ unsigned h[256];
  const int t = threadIdx.x;
  for (int b = 0; b < BATCHES; ++b) {
    int active = (int)ws[WS_ACTIVE + b];       // uniform across block
    if (active) {
      unsigned v = ws[WS_HIST + (b << 8) + t];
      unsigned krem = ws[WS_KREM + b];
      h[t] = v;
      __syncthreads();
      for (int off = 1; off < 256; off <<= 1) {  // Hillis-Steele inclusive scan
        unsigned x = (t >= off) ? h[t - off] : 0u;
        __syncthreads();
        h[t] += x;
        __syncthreads();
      }
      unsigned incl = h[t];
      unsigned excl = incl - v;
      if (v != 0u && krem >= excl && krem < incl) {   // exactly one winner
        unsigned pfx = ws[WS_PREFIX + b] | (((unsigned)t) << shift);
        ws[WS_PREFIX + b] = pfx;
        ws[WS_KREM + b] = krem - excl;
        if (shift == 0) ((float*)ws)[WS_THRESH + b] = ord2f(pfx);
      }
      __syncthreads();
    }
  }
  __syncthreads();
  for (int i = t; i < BATCHES * 256; i += blockDim.x) ws[WS_HIST + i] = 0u;
}

// ---- keep flags (0/1 floats), second half of d_out ----
__global__ void k_keep(const float* __restrict__ pred, const int* __restrict__ bids,
                       const unsigned* __restrict__ ws, float* __restrict__ keep_out, int n) {
  int i = blockIdx.x * blockDim.x + threadIdx.x;
  if (i < n) {
    float th = ((const float*)ws)[WS_THRESH + bids[i]];
    keep_out[i] = (pred[i] > th) ? 1.0f : 0.0f;
  }
}

// ---- streaming prune: async global->LDS pipeline, 512B chunks per wave ----
// PK loads in flight; SLOTS = 2*PK LDS slots so a refill never targets a slot
// whose ds_read could still be outstanding (WAR distance = PK iterations).
#define PK 4                 // async loads in flight per wave
#define SLOTS 8              // LDS slots per wave (power of two)
#define WPB 8                // waves per 256-thread block
#define CHUNK_FLOATS 128     // 512 bytes = 32 lanes * 16B

__global__ void k_prune(const float* __restrict__ fea, const float* __restrict__ keep,
                        float* __restrict__ out, long long nchunks) {
  __shared__ float smem[WPB * SLOTS * CHUNK_FLOATS + CHUNK_FLOATS]; // + dummy slot
  const int lane = threadIdx.x & 31;
  const int wib  = threadIdx.x >> 5;
  const long long wave = (long long)blockIdx.x * WPB + wib;
  const long long nw   = (long long)gridDim.x * WPB;
  const long long cpw  = (nchunks + nw - 1) / nw;
  long long g0 = wave * cpw;
  long long g1 = g0 + cpw;
  if (g0 > nchunks) g0 = nchunks;
  if (g1 > nchunks) g1 = nchunks;

  float* wbase = &smem[wib * (SLOTS * CHUNK_FLOATS)];
  // low 32 bits of a generic pointer to __shared__ == LDS byte offset (aperture rule)
  const unsigned slot0 = (unsigned)(uintptr_t)wbase + (unsigned)lane * 16u;
  const unsigned dummy = (unsigned)(uintptr_t)&smem[WPB * SLOTS * CHUNK_FLOATS] + (unsigned)lane * 16u;
  const float* dummyg = fea + lane * 4;

  // prologue: fill the pipeline (dummies keep ASYNCcnt accounting uniform)
  for (int k = 0; k < PK; ++k) {
    long long g = g0 + k;
    const float* gp = (g < g1) ? (fea + g * CHUNK_FLOATS + lane * 4) : dummyg;
    unsigned loff   = (g < g1) ? (slot0 + (unsigned)k * 512u) : dummy;
    asm volatile("global_load_async_to_lds_b128 %0, %1, off" :: "v"(loff), "v"(gp) : "memory");
  }
  for (long long g = g0; g < g1; ++g) {
    asm volatile("s_wait_asynccnt 0x3" ::: "memory");   // oldest of PK in-flight is done
    int i = (int)(g - g0);
    int s = i & (SLOTS - 1);                            // chunk i lives in slot i%SLOTS
    const float4 v = *(const float4*)(wbase + s * CHUNK_FLOATS + lane * 4);
    float kf = keep[g * 2 + (lane >> 4)];               // 2 points per 512B chunk
    float4 o; o.x = v.x * kf; o.y = v.y * kf; o.z = v.z * kf; o.w = v.w * kf;
    *(float4*)(out + g * CHUNK_FLOATS + lane * 4) = o;
    long long gn = g + PK;                              // refill slot (i+PK)%SLOTS
    int sn = (i + PK) & (SLOTS - 1);                    // read PK iterations ago -> no WAR
    const float* gp = (gn < g1) ? (fea + gn * CHUNK_FLOATS + lane * 4) : dummyg;
    unsigned loff   = (gn < g1) ? (slot0 + (unsigned)sn * 512u) : dummy;
    asm volatile("global_load_async_to_lds_b128 %0, %1, off" :: "v"(loff), "v"(gp) : "memory");
  }
  // leftover dummy loads drain at implicit S_WAIT_IDLE before S_ENDPGM
}

// ---- scalar tail for non-multiple-of-128 totals (not hit for N*64) ----
__global__ void k_tail(const float* __restrict__ fea, const float* __restrict__ keep,
                       float* __restrict__ out, long long start, long long total) {
  long long i = start + (long long)blockIdx.x * blockDim.x + threadIdx.x;
  if (i < total) out[i] = fea[i] * keep[i / FEAC];
}

extern "C" void kernel_launch(void* const* d_in, const int* in_sizes, int n_in,
                              void* d_out, int out_size, void* d_ws, size_t ws_size,
                              hipStream_t stream) {
  const float* fea    = (const float*)d_in[0];
  const float* pred   = (const float*)d_in[1];
  const int*   bids   = (const int*)d_in[2];
  const int*   target = (const int*)d_in[3];
  const int n = in_sizes[1];
  const long long total = (long long)in_sizes[0];
  float* out = (float*)d_out;
  float* keep_out = out + total;
  unsigned* ws = (unsigned*)d_ws;

  k_init<<<1, 256, 0, stream>>>(ws);
  const int nb = (n + 255) / 256;
  k_boundary<<<nb, 256, 0, stream>>>(bids, n, (int*)ws);
  k_setup<<<1, 64, 0, stream>>>(target, ws);
  const int shifts[4] = {24, 16, 8, 0};
  for (int p = 0; p < 4; ++p) {
    k_hist<<<2048, 256, 0, stream>>>(pred, bids, n, ws, shifts[p]);
    k_select<<<1, 256, 0, stream>>>(ws, shifts[p]);
  }
  k_keep<<<nb, 256, 0, stream>>>(pred, bids, ws, keep_out, n);
  const long long nchunks = total / CHUNK_FLOATS;
  if (nchunks > 0) k_prune<<<2048, 256, 0, stream>>>(fea, keep_out, out, nchunks);
  const long long done = nchunks * CHUNK_FLOATS;
  const long long rem = total - done;
  if (rem > 0) {
    int tb = (int)((rem + 255) / 256);
    k_tail<<<tb, 256, 0, stream>>>(fea, keep_out, out, done, total);
  }
}